// dgl_GATConv_39625368273417
// MI455X (gfx1250) — compile-verified
//
#include <hip/hip_runtime.h>
#include <math.h>

typedef __attribute__((ext_vector_type(2))) float v2f;
typedef __attribute__((ext_vector_type(8))) float v8f;

#define GAT_H 8
#define GAT_D 16
#define GAT_F 128   // IN_FEATS == H*D == 128
#define M_PER_BLOCK 32   // two 16-row WMMA tiles per block (100000 % 32 == 0)

// ---------------------------------------------------------------------------
// Kernel 1: ft = feat @ W^T via V_WMMA_F32_16X16X4_F32 (exact f32 GEMM).
// Block = 256 threads (8 waves). Wave w owns N-cols [16w,16w+16) == head w.
// B fragments (W) are hoisted into 64 VGPRs once; wave then runs 2 M-tiles,
// each a pure stream of 32 x {global_load_b64 (A) -> v_wmma}.
// ---------------------------------------------------------------------------
__global__ __launch_bounds__(256) void gat_proj(const float* __restrict__ feat,
                                                const float* __restrict__ W,
                                                float* __restrict__ ft) {
  const int wave = threadIdx.x >> 5;         // 0..7 => head / N tile
  const int lane = threadIdx.x & 31;
  const int o0   = wave * 16;                // N tile base (output col)
  const int r    = lane & 15;
  const int kk   = (lane >> 4) * 2;          // lanes 16-31 hold K+2,K+3

  // B fragment: B[k][o] = W[o][k] -> lane holds W[o0+r][4i+kk .. +1]
  const float* bptr = W + (size_t)(o0 + r) * GAT_F + kk;
  v2f b[32];
#pragma unroll
  for (int i = 0; i < 32; ++i) b[i] = *(const v2f*)(bptr + 4 * i);

  const int n0 = blockIdx.x * M_PER_BLOCK;
#pragma unroll
  for (int t = 0; t < M_PER_BLOCK / 16; ++t) {
    const int mbase = n0 + t * 16;
    // A fragment source: feat[mbase+r][4i+kk .. +1]  (row-major, stride 128)
    const float* aptr = feat + (size_t)(mbase + r) * GAT_F + kk;
    v8f c = {};
#pragma unroll
    for (int i = 0; i < 32; ++i) {
      v2f a = *(const v2f*)(aptr + 4 * i);
      c = __builtin_amdgcn_wmma_f32_16x16x4_f32(false, a, false, b[i],
                                                (short)0, c, false, false);
    }
    // C/D layout: VGPR v, lanes 0-15 -> M=v ; lanes 16-31 -> M=v+8 ; N=lane&15
    const int rowBase = mbase + ((lane >> 4) << 3);
    float* op = ft + (size_t)rowBase * GAT_F + o0 + r;
#pragma unroll
    for (int v = 0; v < 8; ++v) op[(size_t)v * GAT_F] = c[v];
  }
}

// ---------------------------------------------------------------------------
// Kernel 2: per-node attention logits el/er  (one thread per (n,h))
// ---------------------------------------------------------------------------
__global__ void gat_elr(const float* __restrict__ ft,
                        const float* __restrict__ attn_l,
                        const float* __restrict__ attn_r,
                        float* __restrict__ el, float* __restrict__ er, int NH) {
  int t = blockIdx.x * blockDim.x + threadIdx.x;
  if (t >= NH) return;
  int h = t & (GAT_H - 1);
  int n = t >> 3;
  const float4* f  = (const float4*)(ft + (size_t)n * GAT_F + h * GAT_D);
  const float4* al = (const float4*)(attn_l + h * GAT_D);
  const float4* ar = (const float4*)(attn_r + h * GAT_D);
  float sl = 0.f, sr = 0.f;
#pragma unroll
  for (int i = 0; i < 4; ++i) {
    float4 fv = f[i], lv = al[i], rv = ar[i];
    sl += fv.x * lv.x + fv.y * lv.y + fv.z * lv.z + fv.w * lv.w;
    sr += fv.x * rv.x + fv.y * rv.y + fv.z * rv.z + fv.w * rv.w;
  }
  el[t] = sl;
  er[t] = sr;
}

// ---------------------------------------------------------------------------
// Kernel 3: init out = bias (covers zero-in-degree nodes), emax=-inf, denom=0
// ---------------------------------------------------------------------------
__global__ void gat_init(float* __restrict__ out, const float* __restrict__ bias,
                         float* __restrict__ emax, float* __restrict__ denom,
                         int NO, int NH) {
  int t = blockIdx.x * blockDim.x + threadIdx.x;
  if (t < NO) out[t] = bias[t & (GAT_F - 1)];
  if (t < NH) { emax[t] = -INFINITY; denom[t] = 0.f; }
}

__device__ __forceinline__ float lrelu(float x) {
  return x > 0.f ? x : 0.2f * x;
}

// sign-split float atomic max (valid with -inf init)
__device__ __forceinline__ void atomicMaxF(float* addr, float v) {
  if (v >= 0.f)
    atomicMax((int*)addr, __float_as_int(v));
  else
    atomicMin((unsigned int*)addr, __float_as_uint(v));
}

// ---------------------------------------------------------------------------
// Kernel 4: segment max of edge scores into emax[dst]
// ---------------------------------------------------------------------------
__global__ void gat_emax(const int* __restrict__ src, const int* __restrict__ dst,
                         const float* __restrict__ el, const float* __restrict__ er,
                         float* __restrict__ emax, int EH) {
  int t = blockIdx.x * blockDim.x + threadIdx.x;
  if (t >= EH) return;
  int h = t & (GAT_H - 1);
  int e = t >> 3;
  int s = src[e], d = dst[e];
  float v = lrelu(el[s * GAT_H + h] + er[d * GAT_H + h]);
  atomicMaxF(&emax[d * GAT_H + h], v);
}

// ---------------------------------------------------------------------------
// Kernel 5: denom[dst] += exp(e - emax[dst])
// ---------------------------------------------------------------------------
__global__ void gat_denom(const int* __restrict__ src, const int* __restrict__ dst,
                          const float* __restrict__ el, const float* __restrict__ er,
                          const float* __restrict__ emax, float* __restrict__ denom,
                          int EH) {
  int t = blockIdx.x * blockDim.x + threadIdx.x;
  if (t >= EH) return;
  int h = t & (GAT_H - 1);
  int e = t >> 3;
  int s = src[e], d = dst[e];
  float v = lrelu(el[s * GAT_H + h] + er[d * GAT_H + h]);
  float ex = __expf(v - emax[d * GAT_H + h]);
  atomicAdd(&denom[d * GAT_H + h], ex);
}

// ---------------------------------------------------------------------------
// Kernel 6: out[dst] += (exp(e - emax)/denom) * ft[src]   (L2-resident atomics)
// ---------------------------------------------------------------------------
__global__ void gat_agg(const int* __restrict__ src, const int* __restrict__ dst,
                        const float* __restrict__ el, const float* __restrict__ er,
                        const float* __restrict__ emax, const float* __restrict__ denom,
                        const float* __restrict__ ft, float* __restrict__ out, int EH) {
  int t = blockIdx.x * blockDim.x + threadIdx.x;
  if (t >= EH) return;
  int h = t & (GAT_H - 1);
  int e = t >> 3;
  int s = src[e], d = dst[e];
  float v = lrelu(el[s * GAT_H + h] + er[d * GAT_H + h]);
  float a = __expf(v - emax[d * GAT_H + h]) / denom[d * GAT_H + h];
  const float4* fs = (const float4*)(ft + (size_t)s * GAT_F + h * GAT_D);
  float* op = out + (size_t)d * GAT_F + h * GAT_D;
#pragma unroll
  for (int i = 0; i < 4; ++i) {
    float4 f = fs[i];
    atomicAdd(op + 4 * i + 0, a * f.x);
    atomicAdd(op + 4 * i + 1, a * f.y);
    atomicAdd(op + 4 * i + 2, a * f.z);
    atomicAdd(op + 4 * i + 3, a * f.w);
  }
}

extern "C" void kernel_launch(void* const* d_in, const int* in_sizes, int n_in,
                              void* d_out, int out_size, void* d_ws, size_t ws_size,
                              hipStream_t stream) {
  const float* feat   = (const float*)d_in[0];
  const int*   src    = (const int*)d_in[1];
  const int*   dst    = (const int*)d_in[2];
  const float* W      = (const float*)d_in[3];
  const float* attn_l = (const float*)d_in[4];
  const float* attn_r = (const float*)d_in[5];
  const float* bias   = (const float*)d_in[6];
  float*       out    = (float*)d_out;

  const int N = in_sizes[0] / GAT_F;   // 100000
  const int E = in_sizes[1];           // 1600000

  // workspace carve-up (floats): ft[N*128] el[N*8] er[N*8] emax[N*8] denom[N*8]
  float* ft    = (float*)d_ws;
  float* el    = ft + (size_t)N * GAT_F;
  float* er    = el + (size_t)N * GAT_H;
  float* emax  = er + (size_t)N * GAT_H;
  float* denom = emax + (size_t)N * GAT_H;

  const int NH = N * GAT_H;
  const int NO = N * GAT_F;
  const int EH = E * GAT_H;

  gat_proj<<<N / M_PER_BLOCK, 256, 0, stream>>>(feat, W, ft);
  gat_elr<<<(NH + 255) / 256, 256, 0, stream>>>(ft, attn_l, attn_r, el, er, NH);
  gat_init<<<(NO + 255) / 256, 256, 0, stream>>>(out, bias, emax, denom, NO, NH);
  gat_emax<<<(EH + 255) / 256, 256, 0, stream>>>(src, dst, el, er, emax, EH);
  gat_denom<<<(EH + 255) / 256, 256, 0, stream>>>(src, dst, el, er, emax, denom, EH);
  gat_agg<<<(EH + 255) / 256, 256, 0, stream>>>(src, dst, el, er, emax, denom, ft, out, EH);
}